// SparseCrossModalAttentionLayer_74371653697910
// MI455X (gfx1250) — compile-verified
//
#include <hip/hip_runtime.h>
#include <hip/hip_bf16.h>

// Problem constants (fixed by reference setup_inputs)
#define XC      64
#define KDIM    64
#define YS      128
#define BS      8
#define FPS     64
#define PPF     1024
#define NPTS    (BS * FPS * PPF)   // 524288
#define NFRAMES (NPTS / PPF)       // 512
#define CDF_THR 0.9f

typedef __attribute__((ext_vector_type(2))) float v2f;
typedef __attribute__((ext_vector_type(8))) float v8f;

#define TS 68   // LDS row stride (floats): 68 -> bank = (4*row + col) % 64, conflict-free for WMMA gather

// ---------------------------------------------------------------------------
// Kernel 0: bias[b][k] = bc[k] + sum_j y[b][j] * Wy[k][j]      (8 x 64)
// ---------------------------------------------------------------------------
__global__ void bias_kernel(const float* __restrict__ y, const float* __restrict__ Wy,
                            const float* __restrict__ bc, float* __restrict__ biasOut) {
    int t = blockIdx.x * blockDim.x + threadIdx.x;
    if (t < BS * KDIM) {
        int b = t >> 6, k = t & 63;
        float acc = bc[k];
        for (int j = 0; j < YS; ++j) acc += y[b * YS + j] * Wy[k * YS + j];
        biasOut[t] = acc;
    }
}

// ---------------------------------------------------------------------------
// Kernel 1: scores via f32 WMMA.  Block = 256 thr = 8 waves, 16 points/wave.
//   xk(16x64) = x_tile(16x64) @ Wc.T  via V_WMMA_F32_16X16X4_F32
//   score[m]  = sum_n Wa[n]*tanh(xk[m][n] + bias[b][n]) + ba
// A 16x4 layout : lanes 0-15 M=0..15 hold {K=k,k+1}; lanes 16-31 hold {K=k+2,k+3}
// B 4x16 layout : lanes 0-15 N=0..15 hold rows {k,k+1}; lanes 16-31 rows {k+2,k+3}
// C/D 16x16     : vgpr r -> M=r (lanes 0-15) / M=8+r (lanes 16-31), N = lane%16
// ---------------------------------------------------------------------------
__global__ __launch_bounds__(256) void score_kernel(const float* __restrict__ x,
                                                    const float* __restrict__ Wc,
                                                    const float* __restrict__ biasWS,
                                                    const float* __restrict__ Wa,
                                                    const float* __restrict__ ba,
                                                    float* __restrict__ scores) {
    __shared__ float sWc[KDIM * TS];        // 64 rows (out channel) x 64 cols (k), stride 68
    __shared__ float sX[8][16 * TS];        // per-wave 16x64 x tile, stride 68
    __shared__ float sBias[KDIM];
    __shared__ float sWa[KDIM];

    const int t    = threadIdx.x;
    const int wave = t >> 5;
    const int lane = t & 31;
    const int half = lane >> 4;             // 0: lanes 0-15, 1: lanes 16-31
    const int l16  = lane & 15;

    const int blockBase = blockIdx.x * 128;       // 128 points per block, never crosses a frame
    const int b = blockBase >> 16;                // batch id = point / (FPS*PPF)

    // Stage Wc (row n = out channel, col = k)
    for (int i = t; i < KDIM * XC; i += 256) {
        int r = i >> 6, c = i & 63;
        sWc[r * TS + c] = Wc[i];
    }
    if (t < KDIM) { sBias[t] = biasWS[b * KDIM + t]; sWa[t] = Wa[t]; }

    // Stage this wave's x tile: 16 rows x 64 floats, contiguous in memory
    const int ptBase = blockBase + wave * 16;
    const float4* xsrc = (const float4*)(x + (size_t)ptBase * XC);
    float* sx = &sX[wave][0];
    for (int i = lane; i < 256; i += 32) {        // 256 float4 = 1024 floats
        int r = i >> 4;
        int c = (i & 15) << 2;
        float4 v = xsrc[i];
        float* dst = &sx[r * TS + c];
        dst[0] = v.x; dst[1] = v.y; dst[2] = v.z; dst[3] = v.w;
    }
    __syncthreads();

    v8f acc0 = {}, acc1 = {}, acc2 = {}, acc3 = {};
    const float* aRow = &sx[l16 * TS];

#pragma unroll
    for (int k = 0; k < KDIM; k += 4) {
        const int ka = k + half * 2;              // k offset per half-wave (A and B layouts)
        v2f a  = *(const v2f*)&aRow[ka];
        v2f b0 = *(const v2f*)&sWc[(0 * 16 + l16) * TS + ka];
        v2f b1 = *(const v2f*)&sWc[(1 * 16 + l16) * TS + ka];
        v2f b2 = *(const v2f*)&sWc[(2 * 16 + l16) * TS + ka];
        v2f b3 = *(const v2f*)&sWc[(3 * 16 + l16) * TS + ka];
        acc0 = __builtin_amdgcn_wmma_f32_16x16x4_f32(false, a, false, b0, (short)0, acc0, false, false);
        acc1 = __builtin_amdgcn_wmma_f32_16x16x4_f32(false, a, false, b1, (short)0, acc1, false, false);
        acc2 = __builtin_amdgcn_wmma_f32_16x16x4_f32(false, a, false, b2, (short)0, acc2, false, false);
        acc3 = __builtin_amdgcn_wmma_f32_16x16x4_f32(false, a, false, b3, (short)0, acc3, false, false);
    }

    // Fused epilogue: tanh + Wa dot + 16-lane butterfly reduction
    const float wa0 = sWa[ 0 + l16], wa1 = sWa[16 + l16], wa2 = sWa[32 + l16], wa3 = sWa[48 + l16];
    const float bs0 = sBias[ 0 + l16], bs1 = sBias[16 + l16], bs2 = sBias[32 + l16], bs3 = sBias[48 + l16];
    const float baV = ba[0];

#pragma unroll
    for (int r = 0; r < 8; ++r) {
        float p = wa0 * tanhf(acc0[r] + bs0)
                + wa1 * tanhf(acc1[r] + bs1)
                + wa2 * tanhf(acc2[r] + bs2)
                + wa3 * tanhf(acc3[r] + bs3);
        p += __shfl_xor(p, 1, 16);
        p += __shfl_xor(p, 2, 16);
        p += __shfl_xor(p, 4, 16);
        p += __shfl_xor(p, 8, 16);
        if (l16 == 0) scores[ptBase + r + half * 8] = p + baV;
    }
}

// ---------------------------------------------------------------------------
// Kernel 2: per-frame softmax + CDF selection.  Block = 512 thr, 1 frame/block.
// Produces per-point local rank (-1 if dropped) and per-frame keep count,
// reproducing jnp's stable tie ordering exactly.
// ---------------------------------------------------------------------------
__global__ __launch_bounds__(512) void select_kernel(const float* __restrict__ scores,
                                                     int* __restrict__ rankArr,
                                                     int* __restrict__ frameCount) {
    __shared__ float worig[PPF];
    __shared__ float wsort[PPF];
    __shared__ float red[512];
    __shared__ float sThr;
    __shared__ int   sKeep;

    const int f = blockIdx.x;
    const int t = threadIdx.x;
    const float s0 = scores[f * PPF + t];
    const float s1 = scores[f * PPF + t + 512];

    // max
    red[t] = fmaxf(s0, s1);
    __syncthreads();
    for (int off = 256; off > 0; off >>= 1) {
        if (t < off) red[t] = fmaxf(red[t], red[t + off]);
        __syncthreads();
    }
    const float m = red[0];
    __syncthreads();

    // sum of exp
    const float e0 = __expf(s0 - m), e1 = __expf(s1 - m);
    red[t] = e0 + e1;
    __syncthreads();
    for (int off = 256; off > 0; off >>= 1) {
        if (t < off) red[t] += red[t + off];
        __syncthreads();
    }
    const float denom = red[0];
    __syncthreads();

    const float w0 = e0 / denom, w1 = e1 / denom;
    worig[t]       = w0; wsort[t]       = w0;
    worig[t + 512] = w1; wsort[t + 512] = w1;
    __syncthreads();

    // bitonic sort descending (values only)
    for (int k = 2; k <= PPF; k <<= 1) {
        for (int j = k >> 1; j > 0; j >>= 1) {
            for (int idx = t; idx < PPF; idx += 512) {
                int l = idx ^ j;
                if (l > idx) {
                    float a = wsort[idx], bb = wsort[l];
                    bool up = ((idx & k) == 0);            // descending overall
                    if (up ? (a < bb) : (a > bb)) { wsort[idx] = bb; wsort[l] = a; }
                }
            }
            __syncthreads();
        }
    }

    // sequential cumsum (matches jnp.cumsum associativity) -> keep + threshold
    if (t == 0) {
        float c = 0.0f; int keep = PPF;
        for (int i = 0; i < PPF; ++i) {
            c += wsort[i];
            if (c >= CDF_THR) { keep = i + 1; break; }
        }
        sKeep = keep;
        sThr  = wsort[keep - 1];
        // stable membership: all w > thr kept; ties at thr kept in index order
        const float thr = sThr;
        int g = 0;
        for (int i = 0; i < PPF; ++i) if (worig[i] > thr) ++g;
        const int tieAllow = keep - g;
        int cnt = 0, ties = 0;
        for (int i = 0; i < PPF; ++i) {
            float w = worig[i];
            bool kept = false;
            if (w > thr) kept = true;
            else if (w == thr) { kept = (ties < tieAllow); ++ties; }
            rankArr[f * PPF + i] = kept ? cnt++ : -1;
        }
        frameCount[f] = cnt;
    }
}

// ---------------------------------------------------------------------------
// Kernel 3: exclusive scan of per-frame counts (512 values, sequential)
// ---------------------------------------------------------------------------
__global__ void scan_kernel(const int* __restrict__ frameCount, int* __restrict__ frameOff) {
    if (threadIdx.x == 0 && blockIdx.x == 0) {
        int acc = 0;
        for (int f = 0; f < NFRAMES; ++f) { frameOff[f] = acc; acc += frameCount[f]; }
    }
}

// ---------------------------------------------------------------------------
// Kernel 4: streaming compaction.  Block per frame.
// d_out = [ new_x : M*64 floats ][ new_locations : M*3 as float ]
// ---------------------------------------------------------------------------
__global__ __launch_bounds__(256) void compact_kernel(const float* __restrict__ x,
                                                      const int* __restrict__ loc,
                                                      const int* __restrict__ rankArr,
                                                      const int* __restrict__ frameOff,
                                                      float* __restrict__ outX,
                                                      float* __restrict__ outLoc,
                                                      int M) {
    const int f = blockIdx.x;
    const int t = threadIdx.x;
    const int off  = frameOff[f];
    const int base = f * PPF;

    for (int e = t; e < PPF * XC; e += 256) {
        int p = e >> 6;
        int r = rankArr[base + p];
        if (r >= 0) {
            int o = off + r;
            if (o < M) outX[(size_t)o * XC + (e & 63)] = x[(size_t)(base + p) * XC + (e & 63)];
        }
    }
    for (int e = t; e < PPF * 3; e += 256) {
        int p = e / 3, c = e - p * 3;
        int r = rankArr[base + p];
        if (r >= 0) {
            int o = off + r;
            if (o < M) outLoc[(size_t)o * 3 + c] = (float)loc[(size_t)(base + p) * 3 + c];
        }
    }
}

// ---------------------------------------------------------------------------
extern "C" void kernel_launch(void* const* d_in, const int* in_sizes, int n_in,
                              void* d_out, int out_size, void* d_ws, size_t ws_size,
                              hipStream_t stream) {
    const float* x   = (const float*)d_in[0];
    const int*   loc = (const int*)  d_in[1];
    const float* y   = (const float*)d_in[2];
    const float* Wc  = (const float*)d_in[3];
    const float* bc  = (const float*)d_in[4];
    const float* Wy  = (const float*)d_in[5];
    const float* Wa  = (const float*)d_in[6];
    const float* ba  = (const float*)d_in[7];
    (void)in_sizes; (void)n_in; (void)ws_size;

    // workspace layout
    float* wsBias    = (float*)d_ws;                       // 512 floats
    float* wsScores  = wsBias + BS * KDIM;                 // N floats
    int*   wsRank    = (int*)(wsScores + NPTS);            // N ints
    int*   wsFCount  = wsRank + NPTS;                      // 512 ints
    int*   wsFOff    = wsFCount + NFRAMES;                 // 512 ints

    const int M = out_size / (XC + 3);                     // kept rows (64 + 3 elems each)
    float* outX   = (float*)d_out;
    float* outLoc = outX + (size_t)M * XC;

    bias_kernel<<<1, 512, 0, stream>>>(y, Wy, bc, wsBias);
    score_kernel<<<NPTS / 128, 256, 0, stream>>>(x, Wc, wsBias, Wa, ba, wsScores);
    select_kernel<<<NFRAMES, 512, 0, stream>>>(wsScores, wsRank, wsFCount);
    scan_kernel<<<1, 64, 0, stream>>>(wsFCount, wsFOff);
    compact_kernel<<<NFRAMES, 256, 0, stream>>>(x, loc, wsRank, wsFOff, outX, outLoc, M);
}